// CrossLayer_4286377361515
// MI455X (gfx1250) — compile-verified
//
#include <hip/hip_runtime.h>

typedef float v2f __attribute__((ext_vector_type(2)));
typedef float v8f __attribute__((ext_vector_type(8)));

constexpr int Ddim   = 2048;
constexpr int NLAYER = 4;

__global__ __launch_bounds__(256) void dcn_cross_wmma(
    const float* __restrict__ x, const float* __restrict__ W,
    const float* __restrict__ b, float* __restrict__ out, int Brows)
{
    __shared__ float sRed[256];
    __shared__ float sBeta[NLAYER];

    const int tid = threadIdx.x;

    // ---- beta_i = sum_k b[i,k]  (block-redundant; 32KB, L2-resident) ----
    {
        const int layer = tid >> 6;        // 64 threads per layer
        const int off   = tid & 63;
        float p = 0.f;
        #pragma unroll 4
        for (int j = 0; j < Ddim / 64; ++j)
            p += b[layer * Ddim + off + 64 * j];
        sRed[tid] = p;
        __syncthreads();
        if (tid < NLAYER) {
            float s = 0.f;
            for (int j = 0; j < 64; ++j) s += sRed[tid * 64 + j];
            sBeta[tid] = s;
        }
        __syncthreads();
    }

    const int lane = tid & 31;
    const int wave = tid >> 5;
    const int m    = lane & 15;     // row/col index within tile
    const int hi   = lane >> 4;     // K-half selector for A/B layouts

    const int row0 = (blockIdx.x * 8 + wave) * 16;
    if (row0 >= Brows) return;      // wave-uniform; EXEC stays all-ones below

    // ---- Phase 1: u_q = sum_k W[q,k] * x[row0+m,k] via WMMA (q=0..3),
    //               s0  = sum_k x[row0+m,k] scalar-side ----
    // A layout (16x4 f32): lane m holds A[m][2hi..2hi+1]
    // B layout (4x16 f32): lane n holds B[2hi..2hi+1][n]
    const float* xrow  = x + (size_t)(row0 + m) * Ddim + 2 * hi;
    const float  wmask = (m < NLAYER) ? 1.f : 0.f;
    const float* wrow  = W + (size_t)((m < NLAYER) ? m : 0) * Ddim + 2 * hi;

    v8f  acc = {0.f, 0.f, 0.f, 0.f, 0.f, 0.f, 0.f, 0.f};
    float ps = 0.f;
    #pragma unroll 8
    for (int k0 = 0; k0 < Ddim; k0 += 4) {
        v2f bx = *(const v2f*)(xrow + k0);     // x[row m][k0+2hi .. +1]
        v2f wv = *(const v2f*)(wrow + k0);
        v2f a;
        a.x = wv.x * wmask;                    // rows 4..15 of A are zero
        a.y = wv.y * wmask;
        ps += bx.x + bx.y;
        acc = __builtin_amdgcn_wmma_f32_16x16x4_f32(
                  false, a, false, bx, (short)0, acc, false, false);
    }

    // s0: lane m + lane m+16 partials cover all k
    const float s0 = ps + __shfl_xor(ps, 16);
    // D[q][n]: acc[q] lanes 0-15 hold u_q for row n -> gather this lane's row m
    const float u0 = __shfl(acc[0], m);
    const float u1 = __shfl(acc[1], m);
    const float u2 = __shfl(acc[2], m);

    // ---- per-row scalar recurrence: s_{i+1} = s_i*(1+u_i) + beta_i ----
    const float s1 = s0 * (1.f + u0) + sBeta[0];
    const float s2 = s1 * (1.f + u1) + sBeta[1];
    const float s3 = s2 * (1.f + u2) + sBeta[2];

    // Phase-2 A matrix = S (16 rows x 4 layers): already in-lane per layout
    v2f a2;
    a2.x = hi ? s2 : s0;
    a2.y = hi ? s3 : s1;

    // ---- Phase 2: out = x * (1 + S@W) + colsum(b), 16-col chunks ----
    const float* w0 = W + (size_t)(2 * hi)     * Ddim + m;
    const float* w1 = W + (size_t)(2 * hi + 1) * Ddim + m;
    #pragma unroll 2
    for (int n0 = 0; n0 < Ddim; n0 += 16) {
        v2f b2;
        b2.x = w0[n0];                          // W[2hi  ][n0+m]
        b2.y = w1[n0];                          // W[2hi+1][n0+m]
        v8f z = {0.f, 0.f, 0.f, 0.f, 0.f, 0.f, 0.f, 0.f};
        v8f d = __builtin_amdgcn_wmma_f32_16x16x4_f32(
                    false, a2, false, b2, (short)0, z, false, false);

        const int   c  = n0 + m;
        const float bs = b[c] + b[Ddim + c] + b[2 * Ddim + c] + b[3 * Ddim + c];
        #pragma unroll
        for (int v = 0; v < 8; ++v) {
            const size_t idx = (size_t)(row0 + v + 8 * hi) * Ddim + c;
            out[idx] = x[idx] * (1.f + d[v]) + bs;
        }
    }
}

extern "C" void kernel_launch(void* const* d_in, const int* in_sizes, int n_in,
                              void* d_out, int out_size, void* d_ws, size_t ws_size,
                              hipStream_t stream) {
    (void)n_in; (void)out_size; (void)d_ws; (void)ws_size;
    const float* x = (const float*)d_in[0];
    const float* W = (const float*)d_in[1];
    const float* b = (const float*)d_in[2];
    float*     out = (float*)d_out;

    const int Brows = in_sizes[0] / Ddim;          // 8192
    const int grid  = (Brows + 127) / 128;         // 128 rows per block
    dcn_cross_wmma<<<grid, 256, 0, stream>>>(x, W, b, out, Brows);
}